// ExpandEvecs_66013647340022
// MI455X (gfx1250) — compile-verified
//
#include <hip/hip_runtime.h>

typedef __attribute__((ext_vector_type(2))) float v2f;
typedef __attribute__((ext_vector_type(8))) float v8f;

#define MDIM 1024
#define KSTRIDE 32   // evecs last-dim stride (K=32 in memory, we use first 16)
#define LVLS 16

// One wave computes one 16x16 tile of the output for one batch b, producing all
// 16 cumulative levels with V_WMMA_F32_16X16X4_F32 (K processed in 4 chunks of 4,
// per-level masking of the A operand, cumsum carried in the WMMA C accumulator).
__global__ __launch_bounds__(256) void ExpandEvecs_wmma_kernel(
    const float* __restrict__ evecs, float* __restrict__ out) {
  const int lane = threadIdx.x & 31;
  const int wid  = (blockIdx.x << 3) | (threadIdx.x >> 5);  // global wave id
  // wid -> b (4) x mt (64) x nt (64)
  const int b  = wid >> 12;
  const int mt = (wid >> 6) & 63;
  const int nt = wid & 63;

  const int hi = lane >> 4;   // 0: holds k={0,1} of chunk; 1: holds k={2,3}
  const int lp = lane & 15;   // row (A) / col (B) index within tile

  const float* vb   = evecs + (size_t)b * MDIM * KSTRIDE;
  const float* arow = vb + (size_t)(mt * 16 + lp) * KSTRIDE + (hi << 1);
  const float* brow = vb + (size_t)(nt * 16 + lp) * KSTRIDE + (hi << 1);

  // D layout: element r of the v8f -> m = mt*16 + r + 8*hi, n = nt*16 + lp
  float* outb = out + (size_t)b * LVLS * MDIM * MDIM;
  const size_t store_base =
      (size_t)(mt * 16 + (hi << 3)) * MDIM + (size_t)(nt * 16 + lp);

  v8f acc = {};  // cumulative Gram tile through previous K-chunks

#pragma unroll
  for (int j = 0; j < 4; ++j) {  // K-chunk (levels 4j .. 4j+3)
    const v2f a  = *(const v2f*)(arow + 4 * j);
    const v2f bb = *(const v2f*)(brow + 4 * j);
    const float ax_lo = hi ? 0.0f : a.x;  // a.x masked to k<=1
    const float ay_lo = hi ? 0.0f : a.y;  // a.y masked to k<=1 (k=1 vs k=3)

#pragma unroll
    for (int t = 0; t < 4; ++t) {  // level within chunk: keep k <= t
      v2f am;
      am.x = (t < 2) ? ax_lo : a.x;                       // k=0 (lo) / k=2 (hi)
      am.y = (t == 0) ? 0.0f : ((t < 3) ? ay_lo : a.y);   // k=1 (lo) / k=3 (hi)

      v8f d = __builtin_amdgcn_wmma_f32_16x16x4_f32(
          /*neg_a=*/false, am, /*neg_b=*/false, bb,
          /*c_mod=*/(short)0, acc, /*reuse_a=*/false, /*reuse_b=*/false);

      // Stream this level's tile out (write-once data: non-temporal).
      float* o = outb + (size_t)(4 * j + t) * MDIM * MDIM + store_base;
#pragma unroll
      for (int r = 0; r < 8; ++r)
        __builtin_nontemporal_store(d[r], o + (size_t)r * MDIM);

      if (t == 3) acc = d;  // chunk complete: carry cumsum into next chunk
    }
  }
}

extern "C" void kernel_launch(void* const* d_in, const int* in_sizes, int n_in,
                              void* d_out, int out_size, void* d_ws, size_t ws_size,
                              hipStream_t stream) {
  const float* evecs = (const float*)d_in[0];
  float* out = (float*)d_out;
  // 4 (b) * 64 (mt) * 64 (nt) = 16384 waves; 8 waves (256 thr) per block.
  ExpandEvecs_wmma_kernel<<<2048, 256, 0, stream>>>(evecs, out);
}